// AllReduce_57604101374327
// MI455X (gfx1250) — compile-verified
//
#include <hip/hip_runtime.h>
#include <cstdint>

#ifndef __has_builtin
#define __has_builtin(x) 0
#endif

#define TP      4
#define TOKENS  4096
#define HID     8192
#define BLOCK   256
#define VEC     4
#define CHUNKS  (HID / (BLOCK * VEC))   // 8 float4 chunks per thread
#define EPS     1e-6f
#define FP8MAX  448.0f

typedef float f32x4 __attribute__((ext_vector_type(4)));
typedef float f32x2 __attribute__((ext_vector_type(2)));
typedef int   i32x4 __attribute__((ext_vector_type(4)));

// Typed address-space pointers for the async global->LDS builtin
// (signature: v4i AS1*, v4i AS3*, imm offset, imm cpol).
typedef i32x4 __attribute__((address_space(1))) *as1_i32x4_ptr;
typedef i32x4 __attribute__((address_space(3))) *as3_i32x4_ptr;

#if __has_builtin(__builtin_amdgcn_global_load_async_to_lds_b128)
#define HAS_ASYNC_LDS 1
#else
#define HAS_ASYNC_LDS 0
#endif

#if __has_builtin(__builtin_amdgcn_cvt_pk_fp8_f32) && __has_builtin(__builtin_amdgcn_cvt_f32_fp8)
#define HAS_FP8_CVT 1
#else
#define HAS_FP8_CVT 0
#endif

#if __has_builtin(__builtin_amdgcn_cvt_pk_f32_fp8)
#define HAS_FP8_PK_BACK 1
#else
#define HAS_FP8_PK_BACK 0
#endif

// Exact software FP8 e4m3fn RNE round-trip (fallback path):
// clamp to +-448, quantum = 2^(max(floor(log2|x|), -6) - 3), RNE via v_rndne.
__device__ __forceinline__ float fp8_e4m3_roundtrip_sw(float x) {
  x = __builtin_fminf(__builtin_fmaxf(x, -FP8MAX), FP8MAX);
  unsigned abits = __float_as_uint(x) & 0x7fffffffu;
  int e = (int)(abits >> 23) - 127;            // floor(log2|x|) for normals; -127 at x==0
  int q = (e < -6 ? -6 : e) - 3;               // quantum exponent in [-9, 5]
  float quant     = __uint_as_float((unsigned)(q + 127) << 23);
  float inv_quant = __uint_as_float((unsigned)(127 - q) << 23);
  return __builtin_rintf(x * inv_quant) * quant;
}

// Pairwise FP8 e4m3 round-trip: prefer hardware v_cvt_pk_fp8_f32 /
// v_cvt_f32_fp8 (RNE, denorm-correct); clip-then-convert like the reference.
__device__ __forceinline__ f32x2 fp8_roundtrip_pair(float a, float b) {
  f32x2 r;
#if HAS_FP8_CVT
  float ca = __builtin_fminf(__builtin_fmaxf(a, -FP8MAX), FP8MAX);
  float cb = __builtin_fminf(__builtin_fmaxf(b, -FP8MAX), FP8MAX);
  int p = __builtin_amdgcn_cvt_pk_fp8_f32(ca, cb, 0, false);
#if HAS_FP8_PK_BACK
  r = __builtin_amdgcn_cvt_pk_f32_fp8(p, false);
#else
  r.x = __builtin_amdgcn_cvt_f32_fp8(p, 0);
  r.y = __builtin_amdgcn_cvt_f32_fp8(p, 1);
#endif
#else
  r.x = fp8_e4m3_roundtrip_sw(a);
  r.y = fp8_e4m3_roundtrip_sw(b);
#endif
  return r;
}

__global__ __launch_bounds__(BLOCK) void
fused_allreduce_rmsnorm_fp8(const float* __restrict__ in,     // [TP, T, H]
                            const float* __restrict__ resid,  // [T, H]
                            const float* __restrict__ w,      // [H]
                            const float* __restrict__ scale,  // [1]
                            float* __restrict__ out_q,        // [T, H]
                            float* __restrict__ out_r)        // [T, H]
{
  __shared__ float lds_w[HID];            // 32 KB staged weight row
  __shared__ float wg_red[BLOCK / 32];

  const int tid = threadIdx.x;
  const int lane = tid & 31;
  const int wid  = tid >> 5;
  const size_t rowoff = (size_t)blockIdx.x * HID;
  const float s = scale[0];

  // ---- stage the (L2-resident) weight row into LDS, async, overlapped with
  // ---- the main streaming loop below (CDNA5 async global->LDS path).
#if HAS_ASYNC_LDS
  #pragma unroll
  for (int c = 0; c < CHUNKS; ++c) {
    const int eidx = (c * BLOCK + tid) * VEC;
    __builtin_amdgcn_global_load_async_to_lds_b128(
        (as1_i32x4_ptr)(w + eidx),
        (as3_i32x4_ptr)(&lds_w[eidx]),
        0, 0);
  }
#else
  #pragma unroll
  for (int c = 0; c < CHUNKS; ++c) {
    const int eidx = (c * BLOCK + tid) * VEC;
    *(f32x4*)(&lds_w[eidx]) = *(const f32x4*)(w + eidx);
  }
#endif

  // ---- pass 1: all-reduce over TP ranks + residual add, keep row in VGPRs,
  // ---- stream residual_out to HBM with non-temporal b128 stores.
  f32x4 acc[CHUNKS];
  float ss = 0.0f;
  #pragma unroll
  for (int c = 0; c < CHUNKS; ++c) {
    const size_t idx = rowoff + (size_t)((c * BLOCK + tid) * VEC);
    f32x4 v = __builtin_nontemporal_load((const f32x4*)(resid + idx));
    #pragma unroll
    for (int r = 0; r < TP; ++r) {
      f32x4 t = __builtin_nontemporal_load(
          (const f32x4*)(in + (size_t)r * (size_t)TOKENS * HID + idx));
      v += t;
    }
    acc[c] = v;
    ss += v.x * v.x + v.y * v.y + v.z * v.z + v.w * v.w;
    __builtin_nontemporal_store(v, (f32x4*)(out_r + idx));
  }

  // ---- mean-of-squares reduction: wave32 xor-shuffle tree + 8-warp combine.
  #pragma unroll
  for (int off = 16; off > 0; off >>= 1)
    ss += __shfl_xor(ss, off, 32);
  if (lane == 0) wg_red[wid] = ss;
  __syncthreads();
  if (wid == 0) {
    float t = (lane < (BLOCK / 32)) ? wg_red[lane] : 0.0f;
    #pragma unroll
    for (int off = 4; off > 0; off >>= 1)
      t += __shfl_xor(t, off, 32);
    if (lane == 0) wg_red[0] = t;
  }
  __syncthreads();
  const float var = wg_red[0] * (1.0f / (float)HID);
  const float inv = rsqrtf(var + EPS);

  // ---- make sure the async weight staging landed (per-wave counter; each
  // ---- thread reads back exactly the LDS region it issued loads for).
#if HAS_ASYNC_LDS
#if __has_builtin(__builtin_amdgcn_s_wait_asynccnt)
  __builtin_amdgcn_s_wait_asynccnt(0);
#else
  asm volatile("s_wait_asynccnt 0" ::: "memory");
#endif
#endif

  // ---- pass 2: normalize from registers, FP8 e4m3 fake-quant (hardware
  // ---- v_cvt_pk_fp8_f32 when available), NT stores.
  #pragma unroll
  for (int c = 0; c < CHUNKS; ++c) {
    const int eidx = (c * BLOCK + tid) * VEC;
    const size_t idx = rowoff + (size_t)eidx;
    f32x4 wv = *(const f32x4*)(&lds_w[eidx]);
    f32x4 v = acc[c];
    f32x2 lo = fp8_roundtrip_pair(v.x * inv * wv.x * s, v.y * inv * wv.y * s);
    f32x2 hi = fp8_roundtrip_pair(v.z * inv * wv.z * s, v.w * inv * wv.w * s);
    f32x4 o;
    o.x = lo.x;
    o.y = lo.y;
    o.z = hi.x;
    o.w = hi.y;
    __builtin_nontemporal_store(o, (f32x4*)(out_q + idx));
  }
}

extern "C" void kernel_launch(void* const* d_in, const int* in_sizes, int n_in,
                              void* d_out, int out_size, void* d_ws, size_t ws_size,
                              hipStream_t stream) {
  (void)in_sizes; (void)n_in; (void)out_size; (void)d_ws; (void)ws_size;
  const float* input    = (const float*)d_in[0];  // [4, 4096, 8192]
  const float* residual = (const float*)d_in[1];  // [4096, 8192]
  const float* weight   = (const float*)d_in[2];  // [8192]
  const float* scale    = (const float*)d_in[3];  // [1]
  float* out_q = (float*)d_out;                         // quant, [4096, 8192]
  float* out_r = out_q + (size_t)TOKENS * (size_t)HID;  // residual_out

  dim3 grid(TOKENS), block(BLOCK);
  fused_allreduce_rmsnorm_fp8<<<grid, block, 0, stream>>>(
      input, residual, weight, scale, out_q, out_r);
}